// Classifier_76836964926347
// MI455X (gfx1250) — compile-verified
//
#include <hip/hip_runtime.h>
#include <math.h>

// ext-vector types for the f32 WMMA fragments:
// A 16x4 f32 -> 2 VGPRs/lane, B 4x16 f32 -> 2 VGPRs/lane, C/D 16x16 f32 -> 8 VGPRs/lane
typedef float v2f __attribute__((ext_vector_type(2)));
typedef float v8f __attribute__((ext_vector_type(8)));

#define EPS_F 1e-8f

__global__ __launch_bounds__(256)
void classifier_kernel(const float* __restrict__ X,
                       const float* __restrict__ G,   // grp[0], row-major [D][C]
                       float* __restrict__ out)
{
    constexpr int D = 512, C = 100, ROWS = 16;
    constexpr int XS_STRIDE = 516;                    // pad: stride%64 dwords = 4 -> conflict-free cols

    __shared__ float Xs[ROWS][XS_STRIDE];             // 33 KB: 16 rows of X
    __shared__ float dotS[ROWS][113];                 // dot / later cosine-sim
    __shared__ float l1S[ROWS][101];                  // L1 distances
    __shared__ float redA[ROWS][17];
    __shared__ float redB[ROWS][17];
    __shared__ float xnS[ROWS], gnS[112];
    __shared__ float m1S[ROWS], mnS[ROWS], s1S[ROWS], s2S[ROWS];

    const int tid = threadIdx.x;
    const int r0  = blockIdx.x * ROWS;

    __builtin_prefetch(G, 0, 0);                      // global_prefetch_b8 (grp is L2-resident)

    // ---- Phase 1: stage X tile into LDS (float4 -> ds_store_b128) ----
    const float4* X4 = reinterpret_cast<const float4*>(X + (size_t)r0 * D);
    for (int idx = tid; idx < ROWS * (D / 4); idx += 256) {
        int row = idx >> 7;                           // D/4 = 128 float4 per row
        int c4  = idx & 127;
        float4 v = X4[idx];
        *reinterpret_cast<float4*>(&Xs[row][c4 * 4]) = v;
    }
    __syncthreads();

    // ---- Phase 2: row norms (LDS) + class norms (cached global, coalesced) ----
    {
        int row = tid >> 4, seg = tid & 15;
        const float* xr = &Xs[row][seg * 32];
        float s = 0.f;
        #pragma unroll 8
        for (int d = 0; d < 32; ++d) s += xr[d] * xr[d];
        redA[row][seg] = s;
    }
    if (tid < C) {
        float s = 0.f;
        #pragma unroll 4
        for (int d = 0; d < D; ++d) { float gv = G[d * C + tid]; s += gv * gv; }
        gnS[tid] = sqrtf(s);
    }
    __syncthreads();
    if (tid < ROWS) {
        float s = 0.f;
        #pragma unroll
        for (int i = 0; i < 16; ++i) s += redA[tid][i];
        xnS[tid] = sqrtf(s);
    }
    __syncthreads();

    // ---- Phase 3: L1 distances (VALU-bound core). One class per thread,
    //      8-row register accumulators; each g element loaded once, reused 8x. ----
    {
        int c    = tid & 127;
        int half = tid >> 7;
        if (c < C) {
            const int rb = half * 8;
            float acc[8] = {0.f, 0.f, 0.f, 0.f, 0.f, 0.f, 0.f, 0.f};
            #pragma unroll 2
            for (int d = 0; d < D; ++d) {
                float gv = G[d * C + c];
                #pragma unroll
                for (int r = 0; r < 8; ++r)
                    acc[r] += fabsf(Xs[rb + r][d] - gv);
            }
            #pragma unroll
            for (int r = 0; r < 8; ++r) l1S[rb + r][c] = acc[r];
        }
    }

    // ---- Phase 4: dot = X @ g via V_WMMA_F32_16X16X4_F32.
    //      Waves 0..6 each own one 16x16 class tile (C padded to 112).
    //      Branch is wave-uniform -> EXEC all-ones inside (WMMA requirement). ----
    {
        const int wave = tid >> 5;
        const int lane = tid & 31;
        if (wave < 7) {
            const int m   = lane & 15;    // A: row M   / B: col N   / D: col N
            const int hi  = lane >> 4;    // selects K pair {0,1} vs {2,3}
            const int cb  = wave * 16;
            const int col = cb + m;
            const int cc  = (col < C) ? col : (C - 1);  // clamp addr, zero via select
            v8f acc = {0.f, 0.f, 0.f, 0.f, 0.f, 0.f, 0.f, 0.f};
            #pragma unroll 4
            for (int k = 0; k < D; k += 4) {
                const int kk = k + 2 * hi;
                float2 av = *reinterpret_cast<const float2*>(&Xs[m][kk]); // ds_load_b64
                v2f a = { av.x, av.y };
                float bx = G[(size_t)kk * C + cc];
                float by = G[(size_t)(kk + 1) * C + cc];
                v2f b = { (col < C) ? bx : 0.f,
                          (col < C) ? by : 0.f };
                acc = __builtin_amdgcn_wmma_f32_16x16x4_f32(
                        /*neg_a=*/false, a, /*neg_b=*/false, b,
                        /*c_mod=*/(short)0, acc, /*reuse_a=*/false, /*reuse_b=*/false);
            }
            // D layout: VGPR v -> M = v + 8*hi, N = lane&15
            #pragma unroll
            for (int v = 0; v < 8; ++v) {
                int row = v + 8 * hi;
                dotS[row][cb + m] = acc[v];
            }
        }
    }
    __syncthreads();

    // ---- Phase 5: cosine sim + fused softmin/softmax reductions ----
    for (int p = tid; p < ROWS * C; p += 256) {
        int row = p / C, c = p - row * C;
        float cs = dotS[row][c] / fmaxf(xnS[row] * gnS[c], EPS_F);
        dotS[row][c] = cs;
    }
    __syncthreads();
    {
        int row = tid >> 4, t = tid & 15;
        float mx = -3.402823466e38f, mn = 3.402823466e38f;
        for (int c = t; c < C; c += 16) {
            mx = fmaxf(mx, dotS[row][c]);
            mn = fminf(mn, l1S[row][c]);
        }
        redA[row][t] = mx; redB[row][t] = mn;
    }
    __syncthreads();
    if (tid < ROWS) {
        float mx = -3.402823466e38f, mn = 3.402823466e38f;
        #pragma unroll
        for (int i = 0; i < 16; ++i) {
            mx = fmaxf(mx, redA[tid][i]);
            mn = fminf(mn, redB[tid][i]);
        }
        m1S[tid] = mx;           // max cosine sim = confidence & softmax shift
        mnS[tid] = mn;           // min l1 = softmin shift
    }
    __syncthreads();
    {
        int row = tid >> 4, t = tid & 15;
        float m1 = m1S[row], mn = mnS[row];
        float s1 = 0.f, s2 = 0.f;
        for (int c = t; c < C; c += 16) {
            s1 += __expf(dotS[row][c] - m1);
            s2 += __expf(mn - l1S[row][c]);
        }
        redA[row][t] = s1; redB[row][t] = s2;
    }
    __syncthreads();
    if (tid < ROWS) {
        float s1 = 0.f, s2 = 0.f;
        #pragma unroll
        for (int i = 0; i < 16; ++i) { s1 += redA[tid][i]; s2 += redB[tid][i]; }
        s1S[tid] = s1; s2S[tid] = s2;
    }
    __syncthreads();

    // ---- Phase 6: out = confidence * softmax(cs) * softmin(l1), fused single exp ----
    for (int p = tid; p < ROWS * C; p += 256) {
        int row = p / C, c = p - row * C;
        float e   = __expf((dotS[row][c] - m1S[row]) + (mnS[row] - l1S[row][c]));
        float val = m1S[row] * e / (s1S[row] * s2S[row]);
        out[(size_t)(r0 + row) * C + c] = val;
    }
}

extern "C" void kernel_launch(void* const* d_in, const int* in_sizes, int n_in,
                              void* d_out, int out_size, void* d_ws, size_t ws_size,
                              hipStream_t stream) {
    (void)in_sizes; (void)n_in; (void)out_size; (void)d_ws; (void)ws_size;
    const float* X = (const float*)d_in[0];      // [4096, 512]
    const float* G = (const float*)d_in[1];      // [1, 512, 100] -> [D][C]
    float* out = (float*)d_out;                  // [4096, 100]
    dim3 grid(4096 / 16), block(256);
    classifier_kernel<<<grid, block, 0, stream>>>(X, G, out);
}